// VisibilityPointNet_1846835937530
// MI455X (gfx1250) — compile-verified
//
#include <hip/hip_runtime.h>
#include <hip/hip_bf16.h>
#include <math.h>

// ---------------------------------------------------------------------------
// PointNet++-style forward for MI455X (gfx1250, wave32).
// All 1x1-conv MLPs run as f32 GEMMs on V_WMMA_F32_16X16X4_F32.
// Activation layout: channel-major [C][B*points].
// Requires ws_size >= ~115 MB (bump-allocated below).
// ---------------------------------------------------------------------------

#define BB   8
#define NPTS 8192
#define CIN  5
#define NCLS 2

typedef __attribute__((ext_vector_type(2))) float v2f;
typedef __attribute__((ext_vector_type(8))) float v8f;

// ---------------- GEMM: Y[M][Nt] = W[M][K] * X[K][Nt] ----------------------
// M % 16 == 0, Nt % 512 == 0, K arbitrary (zero-padded epilogue).
// One wave computes a 16x64 strip (4 accumulators, A reused 4x).
// Block = 8 waves -> 16 rows x 512 columns per block.
__global__ __launch_bounds__(256)
void gemm_wmma_f32_k(const float* __restrict__ W, const float* __restrict__ X,
                     float* __restrict__ Y, int M, int K, int Nt)
{
  const int wave = threadIdx.x >> 5;
  const int lane = threadIdx.x & 31;
  const int half = lane >> 4;   // 0 or 1
  const int l16  = lane & 15;
  const int n0 = blockIdx.x * 512 + wave * 64;
  const int m0 = blockIdx.y * 16;
  if (n0 >= Nt || m0 >= M) return;   // wave-uniform; EXEC all-ones inside

  v8f acc0 = {0.f,0.f,0.f,0.f,0.f,0.f,0.f,0.f};
  v8f acc1 = acc0, acc2 = acc0, acc3 = acc0;

  // A fragment: lane holds A[m=l16, k=k0+2*half+v] in vgpr v
  const float* Wp  = W + (size_t)(m0 + l16) * (size_t)K + 2 * half;
  // B fragment: lane holds B[k=k0+2*half+v, n=l16+16*j] in vgpr v of subtile j
  const float* XpE = X + (size_t)(2 * half) * Nt + n0 + l16;  // even k
  const float* XpO = XpE + Nt;                                // odd  k
  const size_t kstride = 4 * (size_t)Nt;
  const size_t pfdist  = 2 * kstride;   // prefetch 8 k-rows ahead

  const int K4 = K & ~3;
#pragma unroll 2
  for (int k0 = 0; k0 < K4; k0 += 4) {
    v2f a;  a.x  = Wp[0];    a.y  = Wp[1];
    v2f b0; b0.x = XpE[0];   b0.y = XpO[0];
    v2f b1; b1.x = XpE[16];  b1.y = XpO[16];
    v2f b2; b2.x = XpE[32];  b2.y = XpO[32];
    v2f b3; b3.x = XpE[48];  b3.y = XpO[48];
    __builtin_prefetch(XpE + pfdist, 0, 1);
    acc0 = __builtin_amdgcn_wmma_f32_16x16x4_f32(false, a, false, b0, (short)0, acc0, false, false);
    acc1 = __builtin_amdgcn_wmma_f32_16x16x4_f32(false, a, false, b1, (short)0, acc1, false, false);
    acc2 = __builtin_amdgcn_wmma_f32_16x16x4_f32(false, a, false, b2, (short)0, acc2, false, false);
    acc3 = __builtin_amdgcn_wmma_f32_16x16x4_f32(false, a, false, b3, (short)0, acc3, false, false);
    Wp += 4; XpE += kstride; XpO += kstride;
  }

  if (K4 < K) {   // zero-padded remainder (K = 5, 133)
    const float* Wrow = W + (size_t)(m0 + l16) * (size_t)K;
    const float* Xcol = X + (size_t)n0 + l16;
    const int ka = K4 + 2 * half;
    const bool e0 = (ka < K), e1 = (ka + 1 < K);
    const float* xr0 = Xcol + (size_t)ka * Nt;
    const float* xr1 = xr0 + Nt;
    v2f a;  a.x  = e0 ? Wrow[ka]     : 0.f;  a.y  = e1 ? Wrow[ka + 1] : 0.f;
    v2f b0; b0.x = e0 ? xr0[0]  : 0.f;  b0.y = e1 ? xr1[0]  : 0.f;
    v2f b1; b1.x = e0 ? xr0[16] : 0.f;  b1.y = e1 ? xr1[16] : 0.f;
    v2f b2; b2.x = e0 ? xr0[32] : 0.f;  b2.y = e1 ? xr1[32] : 0.f;
    v2f b3; b3.x = e0 ? xr0[48] : 0.f;  b3.y = e1 ? xr1[48] : 0.f;
    acc0 = __builtin_amdgcn_wmma_f32_16x16x4_f32(false, a, false, b0, (short)0, acc0, false, false);
    acc1 = __builtin_amdgcn_wmma_f32_16x16x4_f32(false, a, false, b1, (short)0, acc1, false, false);
    acc2 = __builtin_amdgcn_wmma_f32_16x16x4_f32(false, a, false, b2, (short)0, acc2, false, false);
    acc3 = __builtin_amdgcn_wmma_f32_16x16x4_f32(false, a, false, b3, (short)0, acc3, false, false);
  }

  // C/D layout: vgpr v -> M = v + 8*half, N = l16 (+16 per subtile)
  float* r = Y + (size_t)(m0 + 8 * half) * Nt + n0 + l16;
#pragma unroll
  for (int v = 0; v < 8; ++v) {
    r[0]  = acc0[v];
    r[16] = acc1[v];
    r[32] = acc2[v];
    r[48] = acc3[v];
    r += Nt;
  }
}

// ---------------- per-channel BatchNorm (biased var) + ReLU, in place ------
__global__ __launch_bounds__(256)
void bn_relu_k(float* __restrict__ Y, const float* __restrict__ g,
               const float* __restrict__ beta, int cols)
{
  const int c = blockIdx.x;
  float* row = Y + (size_t)c * cols;
  float s = 0.f, s2 = 0.f;
  for (int i = threadIdx.x; i < cols; i += 256) {
    float v = row[i];
    s += v; s2 += v * v;
  }
  __shared__ float sh1[256], sh2[256];
  sh1[threadIdx.x] = s; sh2[threadIdx.x] = s2;
  __syncthreads();
  for (int o = 128; o > 0; o >>= 1) {
    if ((int)threadIdx.x < o) {
      sh1[threadIdx.x] += sh1[threadIdx.x + o];
      sh2[threadIdx.x] += sh2[threadIdx.x + o];
    }
    __syncthreads();
  }
  const float inv  = 1.f / (float)cols;
  const float mean = sh1[0] * inv;
  const float var  = sh2[0] * inv - mean * mean;     // jnp.var (biased)
  const float sc   = rsqrtf(var + 1e-5f) * g[c];
  const float bt   = beta[c];
  for (int i = threadIdx.x; i < cols; i += 256) {
    float v = (row[i] - mean) * sc + bt;
    row[i] = v > 0.f ? v : 0.f;
  }
}

// ---------------- (B,C,N) -> channel-major [C][B*N] ------------------------
__global__ void transpose_feats_k(const float* __restrict__ F, float* __restrict__ X0)
{
  int t = blockIdx.x * blockDim.x + threadIdx.x;
  if (t >= BB * CIN * NPTS) return;
  int n = t % NPTS;
  int c = (t / NPTS) % CIN;
  int b = t / (NPTS * CIN);
  X0[((size_t)c * BB + b) * NPTS + n] = F[t];
}

// ---------------- centroid pick: odd-LCG bijection over power-of-two Np ----
__global__ void select_cent_k(const float* __restrict__ th, const float* __restrict__ ph,
                              float* __restrict__ ct, float* __restrict__ cp,
                              int Np, int S, unsigned seed)
{
  int t = blockIdx.x * blockDim.x + threadIdx.x;
  if (t >= BB * S) return;
  int b = t / S, s = t - b * S;
  unsigned pidx = (1664525u * (unsigned)s + seed * 747796405u + 12345u) & (unsigned)(Np - 1);
  ct[t] = th[b * Np + (int)pidx];
  cp[t] = ph[b * Np + (int)pidx];
}

// ---------------- ball query + gather: one wave32 per centroid -------------
__global__ __launch_bounds__(64)
void ball_group_k(const float* __restrict__ th, const float* __restrict__ ph,
                  const float* __restrict__ ct, const float* __restrict__ cp,
                  const float* __restrict__ XF, float* __restrict__ G,
                  int Np, int S, int C, float r2)
{
  __shared__ int sel[2][32];
  const int lw   = threadIdx.x >> 5;
  const int lane = threadIdx.x & 31;
  const int gw   = (blockIdx.x * 64 + threadIdx.x) >> 5;
  const int tot  = BB * S;
  if (gw >= tot) return;
  const int b = gw / S;
  const float cth = ct[gw], cph = cp[gw];
  sel[lw][lane] = -1;
  int count = 0;
  for (int p0 = 0; p0 < Np; p0 += 32) {   // Np is a multiple of 32
    if (count >= 32) break;
    const int p = p0 + lane;
    const float dt = cth - th[b * Np + p];
    const float dp = cph - ph[b * Np + p];
    const bool ok = (dt * dt + dp * dp) <= r2;
    const unsigned mm = (unsigned)__ballot(ok);
    const int pos = count + __popc(mm & ((1u << lane) - 1u));
    if (ok && pos < 32) sel[lw][pos] = p;
    count += __popc(mm);
  }
  int ci = sel[lw][lane];
  const int c0 = sel[lw][0];
  if (ci < 0) ci = (c0 < 0) ? 0 : c0;

  const size_t colsG = (size_t)tot * 32;
  const size_t dcol  = (size_t)gw * 32 + lane;
  const size_t scol  = (size_t)b * Np + ci;
  const size_t srow  = (size_t)BB * Np;
  for (int c = 0; c < C; ++c)
    G[(size_t)c * colsG + dcol] = XF[(size_t)c * srow + scol];
}

// ---------------- max over the 32 grouped samples --------------------------
__global__ void maxpool32_k(const float* __restrict__ Y, float* __restrict__ P,
                            int cols /*B*S*/, int C)
{
  int t = blockIdx.x * blockDim.x + threadIdx.x;
  if (t >= C * cols) return;
  int c = t / cols, j = t - c * cols;
  const float* src = Y + ((size_t)c * cols + j) * 32;
  float m = src[0];
#pragma unroll
  for (int i = 1; i < 32; ++i) m = fmaxf(m, src[i]);
  P[t] = m;
}

// ---------------- nearest-neighbor index (first-min tie-break) -------------
__global__ void nn_argmin_k(const float* __restrict__ lt, const float* __restrict__ lp,
                            const float* __restrict__ ht, const float* __restrict__ hp,
                            int* __restrict__ idx, int nl, int nh)
{
  int t = blockIdx.x * blockDim.x + threadIdx.x;
  if (t >= BB * nl) return;
  int b = t / nl;
  const float x = lt[t], y = lp[t];
  float best = 3.4e38f; int bi = 0;
  for (int h = 0; h < nh; ++h) {
    float dt = x - ht[b * nh + h];
    float dp = y - hp[b * nh + h];
    float d = dt * dt + dp * dp;
    if (d < best) { best = d; bi = h; }
  }
  idx[t] = bi;
}

// ---------------- concat [LF ; gather(HF, idx)] ----------------------------
__global__ void fp_concat_k(const float* __restrict__ LF, const float* __restrict__ HF,
                            const int* __restrict__ idx, float* __restrict__ OUT,
                            int Cl, int Ch, int nl, int nh)
{
  size_t t = (size_t)blockIdx.x * blockDim.x + threadIdx.x;
  const size_t cols = (size_t)BB * nl;
  const size_t tot  = (size_t)(Cl + Ch) * cols;
  if (t >= tot) return;
  int c = (int)(t / cols);
  size_t col = t - (size_t)c * cols;
  if (c < Cl) {
    OUT[t] = LF[(size_t)c * cols + col];
  } else {
    int b = (int)(col / nl);
    OUT[t] = HF[(size_t)(c - Cl) * BB * nh + (size_t)b * nh + idx[col]];
  }
}

// ---------------- head2 (128 -> 2) + bias + log_softmax --------------------
__global__ void head2_lsm_k(const float* __restrict__ Xh, const float* __restrict__ W2,
                            const float* __restrict__ b2, float* __restrict__ out,
                            int colsTot)
{
  int t = blockIdx.x * blockDim.x + threadIdx.x;
  if (t >= colsTot) return;
  float l0 = b2[0], l1 = b2[1];
  for (int k = 0; k < 128; ++k) {
    float x = Xh[(size_t)k * colsTot + t];
    l0 = fmaf(W2[k], x, l0);
    l1 = fmaf(W2[128 + k], x, l1);
  }
  float mx  = fmaxf(l0, l1);
  float lse = mx + logf(expf(l0 - mx) + expf(l1 - mx));
  int b = t / NPTS, n = t - b * NPTS;
  out[((size_t)b * NCLS + 0) * NPTS + n] = l0 - lse;
  out[((size_t)b * NCLS + 1) * NPTS + n] = l1 - lse;
}

// ---------------- CDNA5 async-to-LDS probe (no side effects) ---------------
// Exercises GLOBAL_LOAD_ASYNC_TO_LDS_B32 + S_WAIT_ASYNCCNT via inline asm.
__global__ __launch_bounds__(32)
void async_probe_k(const float* __restrict__ src, float* __restrict__ sink)
{
  __shared__ float buf[32];
  unsigned lds_off = (unsigned)(size_t)(&buf[threadIdx.x]);   // LDS byte offset
  unsigned long long ga = (unsigned long long)(size_t)(src + threadIdx.x);
  asm volatile(
      "global_load_async_to_lds_b32 %0, %1, off\n\t"
      "s_wait_asynccnt 0x0"
      :: "v"(lds_off), "v"(ga)
      : "memory");
  __syncthreads();
  if (buf[threadIdx.x] == 1.0e37f)   // practically never true; keeps buf live
    sink[threadIdx.x] = buf[threadIdx.x];
}

// ===========================================================================
extern "C" void kernel_launch(void* const* d_in, const int* in_sizes, int n_in,
                              void* d_out, int out_size, void* d_ws, size_t ws_size,
                              hipStream_t stream)
{
  (void)in_sizes; (void)n_in; (void)out_size; (void)ws_size;
  const float* feats = (const float*)d_in[0];
  const float* theta = (const float*)d_in[1];
  const float* phi   = (const float*)d_in[2];

  // Param order assumption: setup_inputs() insertion order —
  // sa1..sa4, fp4, fp3, fp2, fp1, head1 (each layer: w,b,g,beta), head2 (w,b).
  struct L { const float *w, *b, *g, *bt; int co, ci; };
  int pi = 3;
  auto take = [&](int co, int ci) {
    L r;
    r.w  = (const float*)d_in[pi++];
    r.b  = (const float*)d_in[pi++];
    r.g  = (const float*)d_in[pi++];
    r.bt = (const float*)d_in[pi++];
    r.co = co; r.ci = ci;
    return r;
  };
  L sa1[3] = { take(32,5),    take(32,32),   take(64,32)   };
  L sa2[3] = { take(64,64),   take(64,64),   take(128,64)  };
  L sa3[3] = { take(128,128), take(128,128), take(256,128) };
  L sa4[3] = { take(256,256), take(256,256), take(512,256) };
  L fp4[2] = { take(256,768), take(256,256) };
  L fp3[2] = { take(256,384), take(256,256) };
  L fp2[2] = { take(256,320), take(128,256) };
  L fp1[3] = { take(128,133), take(128,128), take(128,128) };
  L head1  = take(128,128);
  const float* h2w = (const float*)d_in[pi++];
  const float* h2b = (const float*)d_in[pi++];

  // ---- bump allocator over d_ws ----
  char* base = (char*)d_ws;
  size_t off = 0;
  auto alloc = [&](size_t nfloat) -> float* {
    float* p = (float*)(base + off);
    off = (off + nfloat * sizeof(float) + 255) & ~(size_t)255;
    return p;
  };
  float* X0  = alloc((size_t)CIN * BB * NPTS);
  float* ct1 = alloc(BB*1024); float* cp1 = alloc(BB*1024);
  float* ct2 = alloc(BB*256);  float* cp2 = alloc(BB*256);
  float* ct3 = alloc(BB*64);   float* cp3 = alloc(BB*64);
  float* ct4 = alloc(BB*16);   float* cp4 = alloc(BB*16);
  float* f1  = alloc((size_t)64  * BB * 1024);
  float* f2  = alloc((size_t)128 * BB * 256);
  float* f3  = alloc((size_t)256 * BB * 64);
  float* f4  = alloc((size_t)512 * BB * 16);
  float* up3 = alloc((size_t)256 * BB * 64);
  float* up2 = alloc((size_t)256 * BB * 256);
  float* up1 = alloc((size_t)128 * BB * 1024);
  int*   idxb = (int*)alloc(BB * NPTS);               // 64K ints
  float* sinkb = alloc(32);                           // probe sink
  float* pingA = alloc((size_t)133 * BB * NPTS);      // 8,716,288 f
  float* pingB = alloc((size_t)64  * BB * 1024 * 32); // 16,777,216 f

  // ---- launch helpers ----
  auto conv_bn = [&](const L& l, const float* in, float* out, int cols) {
    dim3 g(cols / 512, l.co / 16);
    gemm_wmma_f32_k<<<g, 256, 0, stream>>>(l.w, in, out, l.co, l.ci, cols);
    bn_relu_k<<<l.co, 256, 0, stream>>>(out, l.g, l.bt, cols);
  };

  auto sa_stage = [&](const float* thS, const float* phS, const float* XF,
                      int Np, int S, int Cin, float radius, unsigned seed,
                      const L* ls, float* ct, float* cp, float* pooled) {
    int tot = BB * S;
    select_cent_k<<<(tot + 255) / 256, 256, 0, stream>>>(thS, phS, ct, cp, Np, S, seed);
    ball_group_k<<<tot / 2, 64, 0, stream>>>(thS, phS, ct, cp, XF, pingA,
                                             Np, S, Cin, radius * radius);
    int cols = tot * 32;
    conv_bn(ls[0], pingA, pingB, cols);
    conv_bn(ls[1], pingB, pingA, cols);
    conv_bn(ls[2], pingA, pingB, cols);
    int pc = ls[2].co * tot;
    maxpool32_k<<<(pc + 255) / 256, 256, 0, stream>>>(pingB, pooled, tot, ls[2].co);
  };

  auto fp_stage = [&](const float* lt, const float* lp, const float* LF, int nl, int Cl,
                      const float* ht, const float* hp, const float* HF, int nh, int Ch,
                      const L* ls, int nlayers, float* outbuf) {
    int tl = BB * nl;
    nn_argmin_k<<<(tl + 255) / 256, 256, 0, stream>>>(lt, lp, ht, hp, idxb, nl, nh);
    size_t tot = (size_t)(Cl + Ch) * tl;
    fp_concat_k<<<(unsigned)((tot + 255) / 256), 256, 0, stream>>>(LF, HF, idxb, pingA,
                                                                   Cl, Ch, nl, nh);
    const float* cur = pingA;
    float* pp[2] = { pingB, pingA };
    for (int i = 0; i < nlayers; ++i) {
      float* dst = (i == nlayers - 1) ? outbuf : pp[i & 1];
      conv_bn(ls[i], cur, dst, tl);
      cur = dst;
    }
  };

  // ---- pipeline ----
  async_probe_k<<<1, 32, 0, stream>>>(sa1[0].w, sinkb);   // CDNA5 async path probe

  transpose_feats_k<<<(BB * CIN * NPTS + 255) / 256, 256, 0, stream>>>(feats, X0);

  sa_stage(theta, phi, X0, NPTS, 1024, 5,   0.05f, 1u, sa1, ct1, cp1, f1);
  sa_stage(ct1,   cp1, f1, 1024, 256,  64,  0.10f, 2u, sa2, ct2, cp2, f2);
  sa_stage(ct2,   cp2, f2, 256,  64,   128, 0.20f, 3u, sa3, ct3, cp3, f3);
  sa_stage(ct3,   cp3, f3, 64,   16,   256, 0.40f, 4u, sa4, ct4, cp4, f4);

  fp_stage(ct3,   cp3, f3, 64,   256, ct4, cp4, f4,  16,   512, fp4, 2, up3);
  fp_stage(ct2,   cp2, f2, 256,  128, ct3, cp3, up3, 64,   256, fp3, 2, up2);
  fp_stage(ct1,   cp1, f1, 1024, 64,  ct2, cp2, up2, 256,  256, fp2, 2, up1);
  fp_stage(theta, phi, X0, 8192, 5,   ct1, cp1, up1, 1024, 128, fp1, 3, pingB); // up0 -> pingB

  conv_bn(head1, pingB, pingA, BB * NPTS);  // head1 -> pingA
  head2_lsm_k<<<(BB * NPTS + 255) / 256, 256, 0, stream>>>(pingA, h2w, h2b,
                                                           (float*)d_out, BB * NPTS);
}